// MultiHeadSelfAttention_25778393711011
// MI455X (gfx1250) — compile-verified
//
#include <hip/hip_runtime.h>
#include <math.h>

// MHA on gfx1250: fp32 WMMA (v_wmma_f32_16x16x4_f32) end-to-end, with
// CDNA5 async global->LDS DMA staging (ASYNCcnt) and double buffering.
// Attention keeps Q A-fragments in registers across the whole key loop.
// B=2, S=2048, D=1024, H=16, HD=64.

typedef __attribute__((ext_vector_type(2))) float v2f;
typedef __attribute__((ext_vector_type(8))) float v8f;

#define B_ 2
#define S_ 2048
#define D_ 1024
#define H_ 16
#define HD_ 64
#define NEGF (-9.0e15f)

__device__ __forceinline__ v8f wmma_f32(v2f a, v2f b, v8f c) {
  // D = A(16x4) * B(4x16) + C(16x16), fp32, wave32.
  return __builtin_amdgcn_wmma_f32_16x16x4_f32(
      /*neg_a=*/false, a, /*neg_b=*/false, b,
      /*c_mod=*/(short)0, c, /*reuse_a=*/false, /*reuse_b=*/false);
}

// ---- CDNA5 async global->LDS copy (16B per lane), tracked by ASYNCcnt ------
__device__ __forceinline__ unsigned lds_u32(const void* p) {
  // shared-space flat address: low 32 bits are the LDS byte offset
  return (unsigned)(uintptr_t)p;
}
__device__ __forceinline__ void async_ld16(unsigned lds, const void* gptr) {
  asm volatile("global_load_async_to_lds_b128 %0, %1, off"
               :: "v"(lds), "v"(gptr) : "memory");
}
__device__ __forceinline__ void async_wait0() {
  asm volatile("s_wait_asynccnt 0x0" ::: "memory");
}

// ---------------------------------------------------------------------------
// GEMM tiling shared by kernels 1 and 3.
// Block = 128 threads (4 waves). Tile = 64x64, K staged in LDS chunks of 32,
// double-buffered via async DMA. LDS strides in bytes are multiples of 16.
// ---------------------------------------------------------------------------
#define AST 36   // As row stride (floats): 144B rows -> 16B aligned chunks
#define BST 68   // Bs row stride (floats): 272B rows -> 16B aligned chunks

__device__ __forceinline__ void stage_gemm_tile(const float* __restrict__ X,
                                                const float* __restrict__ W,
                                                int m0, int n0, int k0,
                                                int K, int N, int tid,
                                                float* As, float* Bs) {
#pragma unroll
  for (int j = 0; j < 4; ++j) {            // A tile 64x32
    int idx = (tid + 128 * j) * 4;
    int r = idx >> 5, c = idx & 31;
    async_ld16(lds_u32(&As[r * AST + c]), X + (size_t)(m0 + r) * K + k0 + c);
  }
#pragma unroll
  for (int j = 0; j < 4; ++j) {            // B tile 32x64
    int idx = (tid + 128 * j) * 4;
    int r = idx >> 6, c = idx & 63;
    async_ld16(lds_u32(&Bs[r * BST + c]), W + (size_t)(k0 + r) * N + n0 + c);
  }
}

__device__ __forceinline__ void gemm_compute_stage(const float* As, const float* Bs,
                                                   int wave, int lo, int hi,
                                                   v8f acc[4]) {
#pragma unroll
  for (int kk = 0; kk < 32; kk += 4) {
    v2f a;
    const float* ap = &As[(wave * 16 + lo) * AST + kk + 2 * hi];
    a.x = ap[0]; a.y = ap[1];
#pragma unroll
    for (int ct = 0; ct < 4; ++ct) {
      v2f b;
      const float* bp = &Bs[(kk + 2 * hi) * BST + ct * 16 + lo];
      b.x = bp[0]; b.y = bp[BST];
      acc[ct] = wmma_f32(a, b, acc[ct]);
    }
  }
}

// ---------------------------------------------------------------------------
// Kernel 1: QKV projection. C[4096,3072] = X[4096,1024] @ Wqkv[1024,3072] + b
// Epilogue scatters into q/k/v as [B,H,S,HD] (bh-major) in workspace.
// ---------------------------------------------------------------------------
__global__ __launch_bounds__(128) void qkv_gemm(const float* __restrict__ X,
                                                const float* __restrict__ W,
                                                const float* __restrict__ bias,
                                                float* __restrict__ qkv_ws) {
  __shared__ float As[2][64 * AST];
  __shared__ float Bs[2][32 * BST];
  const int tid = threadIdx.x;
  const int wave = tid >> 5, lane = tid & 31;
  const int lo = lane & 15, hi = lane >> 4;
  const int n0 = blockIdx.x * 64;          // N = 3072
  const int m0 = blockIdx.y * 64;          // M = 4096
  const int K = D_, N = 3 * D_;

  v8f acc[4] = {};

  stage_gemm_tile(X, W, m0, n0, 0, K, N, tid, As[0], Bs[0]);
  int s = 0;
  for (int k0 = 0; k0 < K; k0 += 32, s ^= 1) {
    async_wait0();
    __syncthreads();
    if (k0 + 32 < K)
      stage_gemm_tile(X, W, m0, n0, k0 + 32, K, N, tid, As[s ^ 1], Bs[s ^ 1]);
    gemm_compute_stage(As[s], Bs[s], wave, lo, hi, acc);
  }

  // Scatter epilogue: col -> (head, q|k|v, jj); q/k/v layout [B*H, S, HD].
  const size_t PLANE = (size_t)B_ * H_ * S_ * HD_;
  float* qp = qkv_ws;
  float* kp = qkv_ws + PLANE;
  float* vp = qkv_ws + 2 * PLANE;
#pragma unroll
  for (int ct = 0; ct < 4; ++ct) {
    int col = n0 + ct * 16 + lo;
    int h = col / 192, rr = col % 192;
    int sel = rr >> 6, jj = rr & 63;
    float* base = (sel == 0) ? qp : ((sel == 1) ? kp : vp);
    float bv = bias[col];
#pragma unroll
    for (int e = 0; e < 8; ++e) {
      int row = m0 + wave * 16 + e + 8 * hi;     // row = b*S + s
      int bb = row >> 11, ss = row & (S_ - 1);
      base[(((size_t)bb * H_ + h) * S_ + ss) * HD_ + jj] = acc[ct][e] + bv;
    }
  }
}

// ---------------------------------------------------------------------------
// Kernel 2: flash attention, causal. grid = (S/64, B*H), block = 128 (4 waves)
// Each wave owns 16 query rows; Q A-fragments live in registers for the whole
// key loop; K/V tiles (16 keys x 64) double-buffered in LDS via async DMA.
// Online softmax; O accumulator = 4 x v8f (16x64).
// ---------------------------------------------------------------------------
#define QST 68
#define KST 68
#define PST 20

__global__ __launch_bounds__(128) void attn_flash(const float* __restrict__ qkv_ws,
                                                  float* __restrict__ ao) {
  __shared__ float Qs[4][16 * QST];
  __shared__ float Ks[2][16 * KST];
  __shared__ float Vs[2][16 * KST];
  __shared__ float Ps[4][16 * PST];

  const int tid = threadIdx.x;
  const int wave = tid >> 5, lane = tid & 31;
  const int lo = lane & 15, hi = lane >> 4;
  const int bh = blockIdx.y;               // b*H + h
  const int q0b = blockIdx.x * 64;
  const int q0 = q0b + wave * 16;

  const size_t PLANE = (size_t)B_ * H_ * S_ * HD_;
  const float* qg = qkv_ws + (size_t)bh * S_ * HD_;
  const float* kg = qkv_ws + PLANE + (size_t)bh * S_ * HD_;
  const float* vg = qkv_ws + 2 * PLANE + (size_t)bh * S_ * HD_;

  // Stage the 64x64 Q tile and K/V tile 0 (async DMA).
#pragma unroll
  for (int j = 0; j < 8; ++j) {
    int idx = (tid + 128 * j) * 4;         // 0..4095
    int r = idx >> 6, c = idx & 63;
    async_ld16(lds_u32(&Qs[r >> 4][(r & 15) * QST + c]),
               qg + (size_t)(q0b + r) * HD_ + c);
  }
#pragma unroll
  for (int j = 0; j < 2; ++j) {
    int idx = (tid + 128 * j) * 4;
    int r = idx >> 6, c = idx & 63;
    async_ld16(lds_u32(&Ks[0][r * KST + c]), kg + (size_t)r * HD_ + c);
    async_ld16(lds_u32(&Vs[0][r * KST + c]), vg + (size_t)r * HD_ + c);
  }
  async_wait0();
  __syncthreads();

  // Hoist Q A-fragments into registers: reused by every key tile.
  v2f qa[16];
#pragma unroll
  for (int d0 = 0; d0 < 64; d0 += 4) {
    const float* ap = &Qs[wave][lo * QST + d0 + 2 * hi];
    qa[d0 >> 2].x = ap[0];
    qa[d0 >> 2].y = ap[1];
  }

  v8f o[4] = {};
  float mrun[8], lrun[8];
#pragma unroll
  for (int e = 0; e < 8; ++e) { mrun[e] = -INFINITY; lrun[e] = 0.0f; }

  const int nkt = (q0b >> 4) + 4;          // causal: keys 0 .. q0b+63
  for (int kt = 0; kt < nkt; ++kt) {
    const int buf = kt & 1;
    if (kt + 1 < nkt) {
#pragma unroll
      for (int j = 0; j < 2; ++j) {        // prefetch next K/V tile (async)
        int idx = (tid + 128 * j) * 4;
        int r = idx >> 6, c = idx & 63;
        async_ld16(lds_u32(&Ks[buf ^ 1][r * KST + c]),
                   kg + (size_t)((kt + 1) * 16 + r) * HD_ + c);
        async_ld16(lds_u32(&Vs[buf ^ 1][r * KST + c]),
                   vg + (size_t)((kt + 1) * 16 + r) * HD_ + c);
      }
    }

    // S = Q * K^T  (contract over HD=64 in 16 WMMA steps; A from registers)
    v8f sc = {};
#pragma unroll
    for (int d0 = 0; d0 < 64; d0 += 4) {
      v2f b;                                // B[d][kcol] = K[kcol][d]
      const float* bp = &Ks[buf][lo * KST + d0 + 2 * hi];
      b.x = bp[0]; b.y = bp[1];
      sc = wmma_f32(qa[d0 >> 2], b, sc);
    }

    // scale, causal mask, online softmax update (row = e + 8*hi, 16 lanes/row)
    int kcol = kt * 16 + lo;
#pragma unroll
    for (int e = 0; e < 8; ++e) {
      int qrow = q0 + e + 8 * hi;
      float sv = sc[e] * 0.125f;            // 1/sqrt(64)
      sv = (kcol <= qrow) ? sv : NEGF;
      float mt = sv;
      mt = fmaxf(mt, __shfl_xor(mt, 1));
      mt = fmaxf(mt, __shfl_xor(mt, 2));
      mt = fmaxf(mt, __shfl_xor(mt, 4));
      mt = fmaxf(mt, __shfl_xor(mt, 8));
      float mnew = fmaxf(mrun[e], mt);
      float p = __expf(sv - mnew);
      float rs = p;
      rs += __shfl_xor(rs, 1);
      rs += __shfl_xor(rs, 2);
      rs += __shfl_xor(rs, 4);
      rs += __shfl_xor(rs, 8);
      float alpha = __expf(mrun[e] - mnew);
      mrun[e] = mnew;
      lrun[e] = lrun[e] * alpha + rs;
#pragma unroll
      for (int ct = 0; ct < 4; ++ct) o[ct][e] *= alpha;
      Ps[wave][(e + 8 * hi) * PST + lo] = p;   // C-layout -> LDS
    }

    // O += P(16x16) * V(16x64): contract over 16 keys in 4 WMMA steps
#pragma unroll
    for (int ks = 0; ks < 16; ks += 4) {
      v2f a;
      const float* ap = &Ps[wave][lo * PST + ks + 2 * hi];
      a.x = ap[0]; a.y = ap[1];
#pragma unroll
      for (int ct = 0; ct < 4; ++ct) {
        v2f b;
        const float* bp = &Vs[buf][(ks + 2 * hi) * KST + ct * 16 + lo];
        b.x = bp[0]; b.y = bp[KST];
        o[ct] = wmma_f32(a, b, o[ct]);
      }
    }

    if (kt + 1 < nkt) {                    // join next tile's DMA
      async_wait0();
      __syncthreads();
    }
  }

  // Finalize: O /= l ; write [b, s, h*64+d] (row-major [4096,1024] for out proj)
  const int bb = bh >> 4, hh = bh & 15;
#pragma unroll
  for (int e = 0; e < 8; ++e) {
    float inv = 1.0f / lrun[e];
    int qrow = q0 + e + 8 * hi;
    float* dst = ao + ((size_t)bb * S_ + qrow) * D_ + hh * HD_;
#pragma unroll
    for (int ct = 0; ct < 4; ++ct)
      dst[ct * 16 + lo] = o[ct][e] * inv;
  }
}

// ---------------------------------------------------------------------------
// Kernel 3: output projection. out[4096,1024] = AO[4096,1024] @ Wout + b_out
// ---------------------------------------------------------------------------
__global__ __launch_bounds__(128) void out_gemm(const float* __restrict__ A,
                                                const float* __restrict__ W,
                                                const float* __restrict__ bias,
                                                float* __restrict__ out) {
  __shared__ float As[2][64 * AST];
  __shared__ float Bs[2][32 * BST];
  const int tid = threadIdx.x;
  const int wave = tid >> 5, lane = tid & 31;
  const int lo = lane & 15, hi = lane >> 4;
  const int n0 = blockIdx.x * 64;          // N = 1024
  const int m0 = blockIdx.y * 64;          // M = 4096
  const int K = D_, N = D_;

  v8f acc[4] = {};

  stage_gemm_tile(A, W, m0, n0, 0, K, N, tid, As[0], Bs[0]);
  int s = 0;
  for (int k0 = 0; k0 < K; k0 += 32, s ^= 1) {
    async_wait0();
    __syncthreads();
    if (k0 + 32 < K)
      stage_gemm_tile(A, W, m0, n0, k0 + 32, K, N, tid, As[s ^ 1], Bs[s ^ 1]);
    gemm_compute_stage(As[s], Bs[s], wave, lo, hi, acc);
  }

#pragma unroll
  for (int ct = 0; ct < 4; ++ct) {
    int col = n0 + ct * 16 + lo;
    float bv = bias[col];
#pragma unroll
    for (int e = 0; e < 8; ++e) {
      int row = m0 + wave * 16 + e + 8 * hi;
      out[(size_t)row * N + col] = acc[ct][e] + bv;
    }
  }
}

// ---------------------------------------------------------------------------
extern "C" void kernel_launch(void* const* d_in, const int* in_sizes, int n_in,
                              void* d_out, int out_size, void* d_ws, size_t ws_size,
                              hipStream_t stream) {
  const float* x    = (const float*)d_in[0];
  // d_in[1] = causal mask (tril), known analytically -> unused
  const float* Wqkv = (const float*)d_in[2];
  const float* bqkv = (const float*)d_in[3];
  const float* Wout = (const float*)d_in[4];
  const float* bout = (const float*)d_in[5];

  float* ws = (float*)d_ws;
  const size_t PLANE = (size_t)B_ * H_ * S_ * HD_;   // 4,194,304 floats
  float* qkv = ws;                // q | k | v, each PLANE floats
  float* ao  = ws + 3 * PLANE;    // attention output [4096,1024]

  qkv_gemm<<<dim3(3 * D_ / 64, B_ * S_ / 64), 128, 0, stream>>>(x, Wqkv, bqkv, qkv);
  attn_flash<<<dim3(S_ / 64, B_ * H_), 128, 0, stream>>>(qkv, ao);
  out_gemm<<<dim3(D_ / 64, B_ * S_ / 64), 128, 0, stream>>>(ao, Wout, bout, (float*)d_out);
}